// CustomLoss_21784074125724
// MI455X (gfx1250) — compile-verified
//
#include <hip/hip_runtime.h>
#include <math.h>

// CDNA5 / gfx1250: wave32. ext vector types for WMMA operands + b128 loads.
typedef __attribute__((ext_vector_type(2))) float v2f;
typedef __attribute__((ext_vector_type(4))) float v4f;
typedef __attribute__((ext_vector_type(8))) float v8f;

// Full 32-lane sum via one V_WMMA_F32_16X16X4_F32.
// A-matrix (16x4 f32) VGPR layout: lane L<16 holds A[M=L][K=0..1] in (a.x,a.y);
// lane L>=16 holds A[M=L-16][K=2..3]. We place acc in a.x and 0 in a.y, B = ones:
//   D[m][n] = acc(lane m) + acc(lane m+16)   (independent of n)
// D layout: VGPR r -> M=r on lanes 0-15, M=r+8 on lanes 16-31. Therefore
//   sum(d[0..7]) on lane 0  == sum of acc over lanes {0..7, 16..23}
//   sum(d[0..7]) on lane 16 == sum of acc over lanes {8..15, 24..31}
// Caller combines the two half-sums (lanes 0 and 16) via an LDS atomic.
__device__ __forceinline__ float wave32_halfsum_wmma(float acc) {
    v2f a; a.x = acc;  a.y = 0.0f;
    v2f b; b.x = 1.0f; b.y = 1.0f;
    v8f c = {};
    // 8 args: (neg_a, A, neg_b, B, c_mod, C, reuse_a, reuse_b)
    v8f d = __builtin_amdgcn_wmma_f32_16x16x4_f32(
        /*neg_a=*/false, a, /*neg_b=*/false, b,
        /*c_mod=*/(short)0, c, /*reuse_a=*/false, /*reuse_b=*/false);
    return d[0] + d[1] + d[2] + d[3] + d[4] + d[5] + d[6] + d[7];
}

__global__ void euclid_init_kernel(float* __restrict__ ws) {
    if (threadIdx.x < 16) ws[threadIdx.x] = 0.0f;
}

// One gridDim.y slice per sample row; grid-stride over the row in float4 units.
// Streaming, bandwidth-bound: non-temporal b128 loads, fp32 fma accumulation.
__global__ void euclid_sumsq_kernel(const float* __restrict__ out,
                                    const float* __restrict__ lab,
                                    float* __restrict__ ws,
                                    int f4_per_row) {
    const int row = blockIdx.y;
    const long long rowBase = (long long)row * (long long)f4_per_row;
    const v4f* __restrict__ o4 = (const v4f*)out + rowBase;
    const v4f* __restrict__ l4 = (const v4f*)lab + rowBase;

    float acc = 0.0f;
    const int stride = gridDim.x * blockDim.x;
    for (int i = blockIdx.x * blockDim.x + threadIdx.x; i < f4_per_row; i += stride) {
        v4f a = __builtin_nontemporal_load(&o4[i]);   // global_load_b128, NT hint
        v4f b = __builtin_nontemporal_load(&l4[i]);
        v4f df = a - b;
        acc = fmaf(df.x, df.x, acc);
        acc = fmaf(df.y, df.y, acc);
        acc = fmaf(df.z, df.z, acc);
        acc = fmaf(df.w, df.w, acc);
    }

    // Wave-level reduction on the matrix pipe (EXEC all-1s here: no divergence).
    float s = wave32_halfsum_wmma(acc);

    __shared__ float lsum;
    if (threadIdx.x == 0) lsum = 0.0f;
    __syncthreads();
    const int lane = threadIdx.x & 31;
    if (lane == 0 || lane == 16) atomicAdd(&lsum, s);   // ds_add_f32
    __syncthreads();
    if (threadIdx.x == 0) atomicAdd(&ws[row], lsum);    // one global atomic per block
}

// Single wave32 block: sqrt each per-sample sum, mean over 16 samples.
__global__ void euclid_finalize_kernel(const float* __restrict__ ws,
                                       float* __restrict__ result) {
    const int lane = threadIdx.x;           // blockDim.x == 32
    float s = 0.0f;
    if (lane < 16) s = sqrtf(ws[lane]);     // reconverges before WMMA
    float p = wave32_halfsum_wmma(s);

    __shared__ float tot;
    if (lane == 0) tot = 0.0f;
    __syncthreads();
    if (lane == 0 || lane == 16) atomicAdd(&tot, p);
    __syncthreads();
    if (lane == 0) result[0] = tot * (1.0f / 16.0f);
}

extern "C" void kernel_launch(void* const* d_in, const int* in_sizes, int n_in,
                              void* d_out, int out_size, void* d_ws, size_t ws_size,
                              hipStream_t stream) {
    const float* out = (const float*)d_in[0];
    const float* lab = (const float*)d_in[1];
    float* res = (float*)d_out;
    float* ws  = (float*)d_ws;   // 16 fp32 partial sums-of-squares

    const int n = in_sizes[0];          // B * D = 16 * 2097152
    const int B = 16;
    const int Dlen = n / B;
    const int f4_per_row = Dlen / 4;    // 524288 float4s per row

    // ~2048 float4s per block-column -> 256 chunks for the reference shape
    int chunks = (f4_per_row + 2047) / 2048;
    if (chunks > 256) chunks = 256;
    if (chunks < 1)   chunks = 1;

    euclid_init_kernel<<<1, 32, 0, stream>>>(ws);

    dim3 grid(chunks, B, 1);
    euclid_sumsq_kernel<<<grid, 256, 0, stream>>>(out, lab, ws, f4_per_row);

    euclid_finalize_kernel<<<1, 32, 0, stream>>>(ws, res);
}